// PolyGNN_53369263620146
// MI455X (gfx1250) — compile-verified
//
#include <hip/hip_runtime.h>

#define HH 64   // hidden width

typedef __attribute__((ext_vector_type(2))) float v2f;
typedef __attribute__((ext_vector_type(8))) float v8f;

// ---------------------------------------------------------------------------
// zero the CSR histogram + allocation counter
// ---------------------------------------------------------------------------
__global__ void k_zero(int* __restrict__ cnt, int* __restrict__ counter, int N) {
    int i = blockIdx.x * blockDim.x + threadIdx.x;
    if (i < N) cnt[i] = 0;
    if (i == 0) *counter = 0;
}

// cnt[dst]++ over all E edges (int atomics)
__global__ void k_count(const int* __restrict__ dst, int* __restrict__ cnt, int E) {
    int e = blockIdx.x * blockDim.x + threadIdx.x;
    if (e < E) atomicAdd(&cnt[dst[e]], 1);
}

// dinv = rsqrt(1 + cnt)   (1 = self loop)
__global__ void k_dinv(const int* __restrict__ cnt, float* __restrict__ dinv, int N) {
    int i = blockIdx.x * blockDim.x + threadIdx.x;
    if (i < N) dinv[i] = rsqrtf(1.0f + (float)cnt[i]);
}

// ---------------------------------------------------------------------------
// CSR row-base allocation: per-block LDS inclusive scan + one global atomic
// per block for the base.
// ---------------------------------------------------------------------------
__global__ __launch_bounds__(256) void k_alloc(const int* __restrict__ cnt,
                                               int* __restrict__ off,
                                               int* __restrict__ cur,
                                               int* __restrict__ counter, int N) {
    __shared__ int s[256];
    __shared__ int base;
    int tid = threadIdx.x;
    int i = blockIdx.x * 256 + tid;
    int c = (i < N) ? cnt[i] : 0;
    int v = c;
    s[tid] = v;
    __syncthreads();
    for (int d = 1; d < 256; d <<= 1) {
        int t = (tid >= d) ? s[tid - d] : 0;
        __syncthreads();
        v += t;
        s[tid] = v;
        __syncthreads();
    }
    if (tid == 255) base = atomicAdd(counter, v);   // v = block total
    __syncthreads();
    if (i < N) {
        int start = base + v - c;                   // exclusive scan + base
        off[i] = start;
        cur[i] = start;
    }
}

// fill adjacency: adj[cur[dst]++] = src
__global__ void k_fill(const int* __restrict__ src, const int* __restrict__ dst,
                       int* __restrict__ cur, int* __restrict__ adj, int E) {
    int e = blockIdx.x * blockDim.x + threadIdx.x;
    if (e >= E) return;
    int p = atomicAdd(&cur[dst[e]], 1);
    adj[p] = src[e];
}

// ---------------------------------------------------------------------------
// Layer 1, fully fused, no float atomics. Wave per node (wave32).
// sum_e norm*(x_s @ W1) == (sum_e norm*x_s) @ W1 since x is 2-wide.
// ---------------------------------------------------------------------------
__global__ __launch_bounds__(256) void k_agg1(const float* __restrict__ x,
                                              const int* __restrict__ adj,
                                              const int* __restrict__ off,
                                              const int* __restrict__ cnt,
                                              const float* __restrict__ dinv,
                                              const float* __restrict__ W1,
                                              const float* __restrict__ b1,
                                              float* __restrict__ out, int N) {
    int wid  = (blockIdx.x * 256 + threadIdx.x) >> 5;   // one wave per node
    int lane = threadIdx.x & 31;
    if (wid >= N) return;                               // wave-uniform exit
    int i = wid;
    float di = dinv[i];
    int o = off[i], c = cnt[i];
    float ax = 0.0f, ay = 0.0f;
    for (int k = lane; k < c; k += 32) {
        int s = adj[o + k];
        float nrm = dinv[s] * di;
        ax += nrm * x[2 * s];
        ay += nrm * x[2 * s + 1];
    }
    for (int d = 16; d; d >>= 1) {                      // wave32 reduce
        ax += __shfl_xor(ax, d, 32);
        ay += __shfl_xor(ay, d, 32);
    }
    float s2 = di * di;                                 // self loop
    ax += s2 * x[2 * i];
    ay += s2 * x[2 * i + 1];
    int f = lane * 2;
    float2 w0 = *(const float2*)(W1 + f);               // W1[0][f..f+1]
    float2 w1 = *(const float2*)(W1 + HH + f);          // W1[1][f..f+1]
    float2 bb = *(const float2*)(b1 + f);
    float2 r;
    r.x = fmaxf(ax * w0.x + ay * w1.x + bb.x, 0.0f);
    r.y = fmaxf(ax * w0.y + ay * w1.y + bb.y, 0.0f);
    *(float2*)(out + (size_t)i * HH + f) = r;
}

// ---------------------------------------------------------------------------
// WMMA fp32 GEMM: C[N,64] = A[N,64] @ W2[64,64]
// W2 staged into LDS with GLOBAL_LOAD_ASYNC_TO_LDS_B128 (ASYNCcnt path),
// B fragments read as ds_load_2addr_b32 pairs (rows k, k+1 of a column).
// Tail-split stores: full tiles take the unguarded coalesced path.
// ---------------------------------------------------------------------------
__global__ __launch_bounds__(256) void k_gemm64(const float* __restrict__ A,
                                                const float* __restrict__ W2,
                                                float* __restrict__ C, int N) {
    __shared__ float w2s[HH * HH];        // W2 row-major copy (untransposed)
    int tid = threadIdx.x;

    // async DMA: 256 lanes x 4 iters x 16B = 16KB, no VGPR staging
    unsigned ldsbase = (unsigned)(uintptr_t)&w2s[0];
#pragma unroll
    for (int r = 0; r < 4; ++r) {
        unsigned byteoff = ((unsigned)tid << 4) + ((unsigned)r << 12);
        unsigned ldsaddr = ldsbase + byteoff;
        const char* gaddr = (const char*)W2 + byteoff;
        asm volatile("global_load_async_to_lds_b128 %0, %1, off"
                     :: "v"(ldsaddr), "v"(gaddr) : "memory");
    }
    asm volatile("s_wait_asynccnt 0" ::: "memory");
    __syncthreads();

    int wave = tid >> 5;                  // wave32 on gfx1250
    int lane = tid & 31;
    int rowBase = blockIdx.x * 128 + wave * 16;
    int row  = rowBase + (lane & 15);
    int rowc = (row < N) ? row : (N - 1); // clamp loads so EXEC stays all-1s
    int kbase = (lane >> 4) << 1;         // lanes 0-15: K+0/1, lanes 16-31: K+2/3
    int col   = lane & 15;
    int mhi   = (lane >> 4) << 3;         // D rows: vgpr r -> M = r (+8 for hi lanes)

    v2f a[16];
    const float* Ap = A + (size_t)rowc * HH + kbase;
#pragma unroll
    for (int ks = 0; ks < 16; ++ks) a[ks] = *(const v2f*)(Ap + ks * 4);

    bool fullTile = (rowBase + 16 <= N);  // wave-uniform

#pragma unroll
    for (int nt = 0; nt < 4; ++nt) {
        v8f c = {};
        const float* Bp = &w2s[kbase * HH + nt * 16 + col];
#pragma unroll
        for (int ks = 0; ks < 16; ++ks) {
            v2f b;
            b.x = Bp[ks * 4 * HH];          // W2[4ks+kbase  ][nt*16+col]
            b.y = Bp[ks * 4 * HH + HH];     // W2[4ks+kbase+1][nt*16+col]
            c = __builtin_amdgcn_wmma_f32_16x16x4_f32(
                    false, a[ks], false, b, (short)0, c, false, false);
        }
        if (fullTile) {
#pragma unroll
            for (int r = 0; r < 8; ++r)
                C[(size_t)(rowBase + r + mhi) * HH + nt * 16 + col] = c[r];
        } else {
#pragma unroll
            for (int r = 0; r < 8; ++r) {
                int grow = rowBase + r + mhi;
                if (grow < N) C[(size_t)grow * HH + nt * 16 + col] = c[r];
            }
        }
    }
}

// ---------------------------------------------------------------------------
// Layer 2 aggregation: wave per node, 2 features/lane -> each incident edge
// is one coalesced 256B row gather of h2lin[src]; lane 0 prefetches the next
// neighbor's row. Fused self loop + bias + ReLU; writes final h.
// ---------------------------------------------------------------------------
__global__ __launch_bounds__(256) void k_agg2(const float* __restrict__ h2,
                                              const int* __restrict__ adj,
                                              const int* __restrict__ off,
                                              const int* __restrict__ cnt,
                                              const float* __restrict__ dinv,
                                              const float* __restrict__ b2,
                                              float* __restrict__ out, int N) {
    int wid  = (blockIdx.x * 256 + threadIdx.x) >> 5;
    int lane = threadIdx.x & 31;
    if (wid >= N) return;
    int i = wid;
    float di = dinv[i];
    int o = off[i], c = cnt[i];
    int f = lane * 2;
    float s2 = di * di;
    float2 hv = *(const float2*)(h2 + (size_t)i * HH + f);
    float ax = s2 * hv.x, ay = s2 * hv.y;
    for (int k = 0; k < c; ++k) {
        int s = adj[o + k];                 // broadcast load (same addr all lanes)
        if (lane == 0 && k + 1 < c) {
            int sn = adj[o + k + 1];
            __builtin_prefetch(h2 + (size_t)sn * HH, 0, 1);
        }
        float nrm = dinv[s] * di;
        float2 v = *(const float2*)(h2 + (size_t)s * HH + f);
        ax += nrm * v.x;
        ay += nrm * v.y;
    }
    float2 bb = *(const float2*)(b2 + f);
    float2 r;
    r.x = fmaxf(ax + bb.x, 0.0f);
    r.y = fmaxf(ay + bb.y, 0.0f);
    *(float2*)(out + (size_t)i * HH + f) = r;
}

// segment starts via binary search (batch is sorted)
__global__ void k_bounds(const int* __restrict__ batch, int* __restrict__ segs,
                         int N, int B) {
    int g = blockIdx.x * blockDim.x + threadIdx.x;
    if (g > B) return;
    if (g == B) { segs[B] = N; return; }
    int lo = 0, hi = N;
    while (lo < hi) {
        int mid = (lo + hi) >> 1;
        if (batch[mid] < g) lo = mid + 1; else hi = mid;
    }
    segs[g] = lo;
}

// ---------------------------------------------------------------------------
// masked mean pool, no atomics: one block per segment, 4 row-streams of 64
// features each (coalesced 256B row reads), LDS reduce, direct stores.
// ---------------------------------------------------------------------------
__global__ __launch_bounds__(256) void k_pool(const float* __restrict__ h,
                                              const int* __restrict__ gid,
                                              const int* __restrict__ segs,
                                              float* __restrict__ psum,
                                              float* __restrict__ pcnt, int B) {
    int g = blockIdx.x;
    int tid = threadIdx.x;
    int f = tid & 63, sub = tid >> 6;           // 4 node-streams x 64 features
    int s0 = segs[g], s1 = segs[g + 1];
    float a0 = 0.0f, a1 = 0.0f, c0 = 0.0f, c1 = 0.0f;
    for (int i = s0 + sub; i < s1; i += 4) {
        int m = gid[i];
        float v = h[(size_t)i * HH + f];
        if (m) { a1 += v; c1 += 1.0f; }
        else   { a0 += v; c0 += 1.0f; }
    }
    __shared__ float red[2][4][HH];
    __shared__ float cred[2][4];
    red[0][sub][f] = a0;
    red[1][sub][f] = a1;
    if (f == 0) { cred[0][sub] = c0; cred[1][sub] = c1; }
    __syncthreads();
    if (sub == 0) {
        float t0 = red[0][0][f] + red[0][1][f] + red[0][2][f] + red[0][3][f];
        float t1 = red[1][0][f] + red[1][1][f] + red[1][2][f] + red[1][3][f];
        psum[(size_t)g * HH + f] = t0;
        psum[((size_t)B + g) * HH + f] = t1;
    }
    if (tid == 0) {
        pcnt[g]     = cred[0][0] + cred[0][1] + cred[0][2] + cred[0][3];
        pcnt[B + g] = cred[1][0] + cred[1][1] + cred[1][2] + cred[1][3];
    }
}

// final FC: out[g][o] = concat(p1,p2)[g] . fcW[:,o] + fcb[o]
__global__ void k_fc(const float* __restrict__ psum, const float* __restrict__ pcnt,
                     const float* __restrict__ fcW, const float* __restrict__ fcb,
                     float* __restrict__ out, int B) {
    int t = blockIdx.x * blockDim.x + threadIdx.x;
    if (t >= B * 7) return;
    int g = t / 7, o = t % 7;
    float c0 = fmaxf(pcnt[g], 1.0f);
    float c1 = fmaxf(pcnt[B + g], 1.0f);
    float acc = fcb[o];
    const float* s0 = psum + (size_t)g * HH;
    const float* s1 = psum + ((size_t)B + g) * HH;
    for (int j = 0; j < HH; ++j) {
        acc += (s0[j] / c0) * fcW[j * 7 + o];
        acc += (s1[j] / c1) * fcW[(HH + j) * 7 + o];
    }
    out[t] = acc;
}

// ---------------------------------------------------------------------------
static inline size_t pad4(size_t n) { return (n + 3) & ~(size_t)3; }

extern "C" void kernel_launch(void* const* d_in, const int* in_sizes, int n_in,
                              void* d_out, int out_size, void* d_ws, size_t ws_size,
                              hipStream_t stream) {
    const float* x    = (const float*)d_in[0];
    const int*   ei   = (const int*)  d_in[1];
    const int*   gid  = (const int*)  d_in[2];
    const int*   bat  = (const int*)  d_in[3];
    const float* W1   = (const float*)d_in[4];
    const float* b1   = (const float*)d_in[5];
    const float* W2   = (const float*)d_in[6];
    const float* b2   = (const float*)d_in[7];
    const float* fcW  = (const float*)d_in[8];
    const float* fcb  = (const float*)d_in[9];
    float* out = (float*)d_out;

    const int N = in_sizes[0] / 2;
    const int E = in_sizes[1] / 2;
    const int B = out_size / 7;
    const int* src = ei;
    const int* dst = ei + E;

    // workspace layout (all regions padded to 4 elements -> 16B alignment)
    char* p = (char*)d_ws;
    int*   cnt     = (int*)p;                 p += pad4(N) * 4;
    int*   off     = (int*)p;                 p += pad4(N) * 4;
    int*   cur     = (int*)p;                 p += pad4(N) * 4;
    int*   segs    = (int*)p;                 p += pad4(B + 1) * 4;
    int*   counter = (int*)p;                 p += pad4(1) * 4;
    int*   adj     = (int*)p;                 p += pad4(E) * 4;
    float* dinv    = (float*)p;               p += pad4(N) * 4;
    float* buf1    = (float*)p;               p += (size_t)N * HH * 4;
    float* buf2    = (float*)p;               p += (size_t)N * HH * 4;
    float* psum    = (float*)p;               p += (size_t)2 * B * HH * 4;
    float* pcnt    = (float*)p;

    const int T = 256;
    int gN    = (N + T - 1) / T;
    int gE    = (E + T - 1) / T;
    int gWave = (int)(((long long)N * 32 + T - 1) / T);   // wave per node
    int gGemm = (N + 127) / 128;
    int gB1   = (B + 1 + T - 1) / T;
    int gFc   = (B * 7 + T - 1) / T;

    k_zero  <<<gN,    T, 0, stream>>>(cnt, counter, N);
    k_count <<<gE,    T, 0, stream>>>(dst, cnt, E);
    k_dinv  <<<gN,    T, 0, stream>>>(cnt, dinv, N);
    k_alloc <<<gN,    T, 0, stream>>>(cnt, off, cur, counter, N);
    k_fill  <<<gE,    T, 0, stream>>>(src, dst, cur, adj, E);
    k_agg1  <<<gWave, T, 0, stream>>>(x, adj, off, cnt, dinv, W1, b1, buf1, N);
    k_gemm64<<<gGemm, T, 0, stream>>>(buf1, W2, buf2, N);
    k_agg2  <<<gWave, T, 0, stream>>>(buf2, adj, off, cnt, dinv, b2, buf1, N);
    k_bounds<<<gB1,   T, 0, stream>>>(bat, segs, N, B);
    k_pool  <<<B,     T, 0, stream>>>(buf1, gid, segs, psum, pcnt, B);
    k_fc    <<<gFc,   T, 0, stream>>>(psum, pcnt, fcW, fcb, out, B);
}